// GraphSAGE_25220047962465
// MI455X (gfx1250) — compile-verified
//
#include <hip/hip_runtime.h>

// ---------------------------------------------------------------------------
// GraphSAGE (PyG SAGEConv mean-aggr, normalize=True) x3 + ReLU, fp32.
// N=10000 nodes, E=160000 edges, dims 512 -> 512 -> 512 -> 256.
// Matrix math runs on V_WMMA_F32_16X16X4_F32 (true fp32 WMMA, matches ref).
// ---------------------------------------------------------------------------

#define SAGE_N 10000
#define SAGE_E 160000

typedef float v2f __attribute__((ext_vector_type(2)));
typedef float v8f __attribute__((ext_vector_type(8)));

// -------------------------------- utility kernels --------------------------

__global__ void sage_zero_u32(unsigned* __restrict__ p, int n) {
    int i = blockIdx.x * blockDim.x + threadIdx.x;
    if (i < n) p[i] = 0u;
}

__global__ void sage_zero_f32(float* __restrict__ p, long long n) {
    long long i = (long long)blockIdx.x * blockDim.x + threadIdx.x;
    if (i < n) p[i] = 0.0f;
}

__global__ void sage_count(const int* __restrict__ dst, unsigned* __restrict__ cnt, int e) {
    int i = blockIdx.x * blockDim.x + threadIdx.x;
    if (i < e) atomicAdd(&cnt[dst[i]], 1u);
}

__global__ void sage_invcnt(const unsigned* __restrict__ cnt, float* __restrict__ inv, int n) {
    int i = blockIdx.x * blockDim.x + threadIdx.x;
    if (i < n) {
        unsigned c = cnt[i];
        inv[i] = 1.0f / (float)(c ? c : 1u);
    }
}

// scatter-add: Msum[dst, :] += x[src, :]
// grid = (E, DIN/256); src/dst are block-uniform -> scalar loads; feature
// index coalesced across lanes -> fully coalesced vmem + f32 L2 atomics.
template <int DIN>
__global__ void sage_scatter(const float* __restrict__ x,
                             const int* __restrict__ esrc,
                             const int* __restrict__ edst,
                             float* __restrict__ Msum) {
    int e = blockIdx.x;
    int f = blockIdx.y * blockDim.x + threadIdx.x;
    int s = esrc[e];
    int d = edst[e];
    atomicAdd(&Msum[(size_t)d * DIN + f], x[(size_t)s * DIN + f]);
}

// ------------------------- fused GEMM + normalize layer --------------------
// out[i,:] = relu( l2norm( (Msum[i,:]*invcnt[i]) @ Wl.T + b + x[i,:] @ Wr.T ) )
//
// Workgroup: 256 threads = 8 wave32, tiled 2 (M) x 4 (N); owns a 32-row x
// DOUT strip so the row norm is finished in-kernel.
// Each wave: 16 rows x (DOUT/4) cols = TN 16x16 f32 WMMA tiles, K = 2*DIN
// via V_WMMA_F32_16X16X4_F32.
template <int DIN, int DOUT>
__global__ __launch_bounds__(256, 1)
void sage_layer(const float* __restrict__ Msum,
                const float* __restrict__ xin,
                const float* __restrict__ invcnt,
                const float* __restrict__ Wl,
                const float* __restrict__ bias,
                const float* __restrict__ Wr,
                float* __restrict__ out,
                int nrows) {
    constexpr int TN = DOUT / 64;   // WMMA tiles per wave along N
    __shared__ float s_sq[32];

    const int tid  = threadIdx.x;
    const int lane = tid & 31;
    const int wave = tid >> 5;
    const int wm   = wave >> 2;     // 0..1 : 16-row group
    const int wn   = wave & 3;      // 0..3 : DOUT/4 column group
    const int l16  = lane & 15;
    const int hi   = lane >> 4;     // 0/1 : K sub-pair per ISA A/B layout

    if (tid < 32) s_sq[tid] = 0.0f;
    __syncthreads();

    const int  rowg   = blockIdx.x * 32 + wm * 16 + l16;  // A-fragment row
    const bool rvalid = rowg < nrows;
    const int  rowc   = rvalid ? rowg : 0;

    v8f acc[TN];
    #pragma unroll
    for (int t = 0; t < TN; ++t)
        #pragma unroll
        for (int g = 0; g < 8; ++g) acc[t][g] = 0.0f;

    // phase 0: A = mean (Msum * invcnt), B = Wl ; phase 1: A = x, B = Wr
    #pragma unroll
    for (int phase = 0; phase < 2; ++phase) {
        const float* Ab  = phase ? xin : Msum;
        const float* Bb  = phase ? Wr  : Wl;
        const float  asc = phase ? (rvalid ? 1.0f : 0.0f)
                                 : (rvalid ? invcnt[rowc] : 0.0f);
        // lane holds A[row, k0 + 2*hi + {0,1}] -> one 8B load per fragment
        const float* arow = Ab + (size_t)rowc * DIN + 2 * hi;
        const float* brow[TN];
        #pragma unroll
        for (int t = 0; t < TN; ++t) {
            int col = wn * (16 * TN) + t * 16 + l16;
            // B[k, n] = W[n, k]; W row-major over k -> contiguous float2
            brow[t] = Bb + (size_t)col * DIN + 2 * hi;
        }
        for (int k0 = 0; k0 < DIN; k0 += 4) {
            v2f a = *reinterpret_cast<const v2f*>(arow + k0);
            a *= asc;                          // fold mean-division into A
            #pragma unroll
            for (int t = 0; t < TN; ++t) {
                v2f b = *reinterpret_cast<const v2f*>(brow[t] + k0);
                acc[t] = __builtin_amdgcn_wmma_f32_16x16x4_f32(
                    /*neg_a=*/false, a, /*neg_b=*/false, b,
                    /*c_mod=*/(short)0, acc[t],
                    /*reuse_a=*/false, /*reuse_b=*/false);
            }
        }
    }

    // + bias (same value for all 8 rows a lane holds in a tile column)
    #pragma unroll
    for (int t = 0; t < TN; ++t) {
        float bb = bias[wn * (16 * TN) + t * 16 + l16];
        #pragma unroll
        for (int g = 0; g < 8; ++g) acc[t][g] += bb;
    }

    // row sum-of-squares: C layout -> VGPR g = row (g + 8*hi), lane&15 = col.
    // Reduce over the 16 lanes of each half, then over the 4 wn-waves via LDS.
    #pragma unroll
    for (int g = 0; g < 8; ++g) {
        float s = 0.0f;
        #pragma unroll
        for (int t = 0; t < TN; ++t) s += acc[t][g] * acc[t][g];
        s += __shfl_xor(s, 1, 32);
        s += __shfl_xor(s, 2, 32);
        s += __shfl_xor(s, 4, 32);
        s += __shfl_xor(s, 8, 32);
        if (l16 == 0) atomicAdd(&s_sq[wm * 16 + hi * 8 + g], s);
    }
    __syncthreads();

    // normalize + relu + store
    #pragma unroll
    for (int g = 0; g < 8; ++g) {
        int rloc = wm * 16 + hi * 8 + g;
        int rg   = blockIdx.x * 32 + rloc;
        if (rg < nrows) {
            float inv = 1.0f / fmaxf(sqrtf(s_sq[rloc]), 1e-12f);
            #pragma unroll
            for (int t = 0; t < TN; ++t) {
                int col = wn * (16 * TN) + t * 16 + l16;
                out[(size_t)rg * DOUT + col] = fmaxf(acc[t][g] * inv, 0.0f);
            }
        }
    }
}

// ------------------------------- launcher ----------------------------------

static inline size_t align256(size_t x) { return (x + 255) & ~(size_t)255; }

extern "C" void kernel_launch(void* const* d_in, const int* in_sizes, int n_in,
                              void* d_out, int out_size, void* d_ws, size_t ws_size,
                              hipStream_t stream) {
    (void)in_sizes; (void)n_in; (void)out_size; (void)ws_size;
    const int   N = SAGE_N;
    const int   E = SAGE_E;

    const float* x    = (const float*)d_in[0];
    const int*   eidx = (const int*)d_in[1];   // [2, E] int32 (JAX x64 off)
    const float* Wl0  = (const float*)d_in[2];
    const float* b0   = (const float*)d_in[3];
    const float* Wr0  = (const float*)d_in[4];
    const float* Wl1  = (const float*)d_in[5];
    const float* b1   = (const float*)d_in[6];
    const float* Wr1  = (const float*)d_in[7];
    const float* Wl2  = (const float*)d_in[8];
    const float* b2   = (const float*)d_in[9];
    const float* Wr2  = (const float*)d_in[10];
    float*       outp = (float*)d_out;

    const int* esrc = eidx;
    const int* edst = eidx + E;

    // workspace carve-up
    char*  w      = (char*)d_ws;
    size_t off    = 0;
    unsigned* cnt = (unsigned*)(w + off); off = align256(off + (size_t)N * 4);
    float* invc   = (float*)(w + off);    off = align256(off + (size_t)N * 4);
    float* Msum   = (float*)(w + off);    off = align256(off + (size_t)N * 512 * 4);
    float* bufA   = (float*)(w + off);    off = align256(off + (size_t)N * 512 * 4);
    float* bufB   = (float*)(w + off);

    const int B256 = 256;
    const long long F512 = (long long)N * 512;

    // degree counts (shared by all layers)
    sage_zero_u32<<<(N + B256 - 1) / B256, B256, 0, stream>>>(cnt, N);
    sage_count<<<(E + B256 - 1) / B256, B256, 0, stream>>>(edst, cnt, E);
    sage_invcnt<<<(N + B256 - 1) / B256, B256, 0, stream>>>(cnt, invc, N);

    dim3 sgrid((unsigned)E, 2);                  // E blocks x (512/256) feature slabs
    const int ggrid = (N + 31) / 32;             // 313 strips of 32 rows

    // ---- layer 0: x(512) -> bufA(512)
    sage_zero_f32<<<(unsigned)((F512 + B256 - 1) / B256), B256, 0, stream>>>(Msum, F512);
    sage_scatter<512><<<sgrid, B256, 0, stream>>>(x, esrc, edst, Msum);
    sage_layer<512, 512><<<ggrid, B256, 0, stream>>>(Msum, x, invc, Wl0, b0, Wr0, bufA, N);

    // ---- layer 1: bufA(512) -> bufB(512)
    sage_zero_f32<<<(unsigned)((F512 + B256 - 1) / B256), B256, 0, stream>>>(Msum, F512);
    sage_scatter<512><<<sgrid, B256, 0, stream>>>(bufA, esrc, edst, Msum);
    sage_layer<512, 512><<<ggrid, B256, 0, stream>>>(Msum, bufA, invc, Wl1, b1, Wr1, bufB, N);

    // ---- layer 2: bufB(512) -> out(256)
    sage_zero_f32<<<(unsigned)((F512 + B256 - 1) / B256), B256, 0, stream>>>(Msum, F512);
    sage_scatter<512><<<sgrid, B256, 0, stream>>>(bufB, esrc, edst, Msum);
    sage_layer<512, 256><<<ggrid, B256, 0, stream>>>(Msum, bufB, invc, Wl2, b2, Wr2, outp, N);
}